// Generator3DLUT_72928544686086
// MI455X (gfx1250) — compile-verified
//
#include <hip/hip_runtime.h>

// Trilinear 3D LUT (DIM=33) applied to [8,3,1024,1024] f32 images.
// Memory-bound: ~201 MB HBM traffic -> ~8.6 us floor at 23.3 TB/s.
// Strategy:
//   1) repack LUT [3][33][33][33] -> AoS float4 (c0,c1,c2,0) per lattice point
//      in d_ws (575 KB): each corner gather is one aligned b128 load and the
//      r/r+1 corner pairs are contiguous 32B spans.
//   2) apply kernel: 4 pixels/lane (float4 b128), non-temporal loads of x and
//      non-temporal stores of out (CDNA5 TH=NT cache hints) so the 200 MB
//      stream never evicts the packed LUT from WGP$/L2. 2D grid (y = batch)
//      keeps all indexing 32-bit (no div/mod), and the interpolation chain is
//      written on 3 explicit channels (42 FMA-ops/pixel).

#define DIM   33
#define DIM2  (DIM * DIM)          // 1089
#define NLAT  (DIM * DIM * DIM)    // 35937

typedef __attribute__((ext_vector_type(4))) float v4f;

struct f3 { float x, y, z; };

static __device__ __forceinline__ int clamp31(int v) {
    v = v < 0 ? 0 : v;
    return v > (DIM - 2) ? (DIM - 2) : v;
}

static __device__ __forceinline__ f3 lerp3v(const v4f a, const v4f b,
                                            float t, float tm) {
    f3 r;
    r.x = a.x * tm + b.x * t;
    r.y = a.y * tm + b.y * t;
    r.z = a.z * tm + b.z * t;
    return r;
}

static __device__ __forceinline__ f3 lerp3(const f3 a, const f3 b,
                                           float t, float tm) {
    f3 r;
    r.x = a.x * tm + b.x * t;
    r.y = a.y * tm + b.y * t;
    r.z = a.z * tm + b.z * t;
    return r;
}

// ---------------------------------------------------------------------------
// Kernel 1: repack LUT planes into AoS float4 lattice in workspace.
// ---------------------------------------------------------------------------
__global__ __launch_bounds__(256) void lut_repack_kernel(
    const float* __restrict__ lut, v4f* __restrict__ packed) {
    int i = blockIdx.x * blockDim.x + threadIdx.x;
    if (i < NLAT) {
        v4f p;
        p.x = lut[i];                 // channel 0
        p.y = lut[NLAT + i];          // channel 1
        p.z = lut[2 * NLAT + i];      // channel 2
        p.w = 0.0f;
        packed[i] = p;
    }
}

// ---------------------------------------------------------------------------
// Kernel 2: apply trilinear interpolation, packed-LUT path.
// grid: (HW/4/256, B); each thread handles 4 consecutive pixels of image
// blockIdx.y. All indexing is 32-bit (tensor < 2^31 elements).
// ---------------------------------------------------------------------------
__global__ __launch_bounds__(256) void lut_apply_packed_kernel(
    const float* __restrict__ x, const v4f* __restrict__ P,
    float* __restrict__ out, int HW) {

    const float inv_bin = 32.0f / 1.0001f;   // 1/binsize, binsize = 1.0001/(DIM-1)

    const int q = blockIdx.x * blockDim.x + threadIdx.x;   // quad within image
    const int p4 = q << 2;                                 // pixel index
    if (p4 >= HW) return;
    const int base = blockIdx.y * 3 * HW + p4;

    // Streamed, read-once: non-temporal b128 loads (th:TH_LOAD_NT).
    v4f r4 = __builtin_nontemporal_load((const v4f*)(x + base));
    v4f g4 = __builtin_nontemporal_load((const v4f*)(x + base + HW));
    v4f b4 = __builtin_nontemporal_load((const v4f*)(x + base + 2 * HW));

    v4f outR, outG, outB;

#pragma unroll
    for (int k = 0; k < 4; ++k) {
        float rs = r4[k] * inv_bin;
        float gs = g4[k] * inv_bin;
        float bs = b4[k] * inv_bin;

        int ri = clamp31((int)floorf(rs));
        int gi = clamp31((int)floorf(gs));
        int bi = clamp31((int)floorf(bs));

        float rd = rs - (float)ri;
        float gd = gs - (float)gi;
        float bd = bs - (float)bi;
        float rm = 1.0f - rd, gm = 1.0f - gd, bm = 1.0f - bd;

        const v4f* p = P + (bi * DIM2 + gi * DIM + ri);

        // 8 corners: 4 contiguous 32B pairs at immediate offsets from one
        // base -> temporal b128 loads keep the packed LUT hot in WGP$/L2.
        v4f c000 = p[0];               v4f c100 = p[1];
        v4f c010 = p[DIM];             v4f c110 = p[DIM + 1];
        v4f c001 = p[DIM2];            v4f c101 = p[DIM2 + 1];
        v4f c011 = p[DIM2 + DIM];      v4f c111 = p[DIM2 + DIM + 1];

        // Factored trilinear on 3 explicit channels (pad lane never computed).
        f3 ar = lerp3v(c000, c100, rd, rm);
        f3 br = lerp3v(c010, c110, rd, rm);
        f3 cr = lerp3v(c001, c101, rd, rm);
        f3 dr = lerp3v(c011, c111, rd, rm);
        f3 eg = lerp3(ar, br, gd, gm);
        f3 fg = lerp3(cr, dr, gd, gm);
        f3 o  = lerp3(eg, fg, bd, bm);

        outR[k] = o.x;
        outG[k] = o.y;
        outB[k] = o.z;
    }

    // Write-once output: non-temporal b128 stores (th:TH_STORE_NT).
    __builtin_nontemporal_store(outR, (v4f*)(out + base));
    __builtin_nontemporal_store(outG, (v4f*)(out + base + HW));
    __builtin_nontemporal_store(outB, (v4f*)(out + base + 2 * HW));
}

// ---------------------------------------------------------------------------
// Fallback: gather directly from the original [3][33][33][33] layout
// (used only if the workspace cannot hold the 575 KB packed table).
// ---------------------------------------------------------------------------
__global__ __launch_bounds__(256) void lut_apply_direct_kernel(
    const float* __restrict__ x, const float* __restrict__ lut,
    float* __restrict__ out, int HW) {

    const float inv_bin = 32.0f / 1.0001f;

    const int q = blockIdx.x * blockDim.x + threadIdx.x;
    const int p4 = q << 2;
    if (p4 >= HW) return;
    const int base = blockIdx.y * 3 * HW + p4;

    v4f r4 = __builtin_nontemporal_load((const v4f*)(x + base));
    v4f g4 = __builtin_nontemporal_load((const v4f*)(x + base + HW));
    v4f b4 = __builtin_nontemporal_load((const v4f*)(x + base + 2 * HW));

    v4f outC[3];

#pragma unroll
    for (int k = 0; k < 4; ++k) {
        float rs = r4[k] * inv_bin;
        float gs = g4[k] * inv_bin;
        float bs = b4[k] * inv_bin;

        int ri = clamp31((int)floorf(rs));
        int gi = clamp31((int)floorf(gs));
        int bi = clamp31((int)floorf(bs));

        float rd = rs - (float)ri;
        float gd = gs - (float)gi;
        float bd = bs - (float)bi;
        float rm = 1.0f - rd, gm = 1.0f - gd, bm = 1.0f - bd;

        int idx = bi * DIM2 + gi * DIM + ri;

#pragma unroll
        for (int c = 0; c < 3; ++c) {
            const float* L = lut + c * NLAT + idx;
            float a = L[0]            * rm + L[1]              * rd;
            float b = L[DIM]          * rm + L[DIM + 1]        * rd;
            float e = a * gm + b * gd;
            float cc = L[DIM2]        * rm + L[DIM2 + 1]       * rd;
            float d = L[DIM2 + DIM]   * rm + L[DIM2 + DIM + 1] * rd;
            float f = cc * gm + d * gd;
            outC[c][k] = e * bm + f * bd;
        }
    }

    __builtin_nontemporal_store(outC[0], (v4f*)(out + base));
    __builtin_nontemporal_store(outC[1], (v4f*)(out + base + HW));
    __builtin_nontemporal_store(outC[2], (v4f*)(out + base + 2 * HW));
}

// ---------------------------------------------------------------------------
extern "C" void kernel_launch(void* const* d_in, const int* in_sizes, int n_in,
                              void* d_out, int out_size, void* d_ws, size_t ws_size,
                              hipStream_t stream) {
    const float* x   = (const float*)d_in[0];   // [B,3,H,W] f32
    const float* lut = (const float*)d_in[1];   // [3,33,33,33] f32
    float* out = (float*)d_out;

    const int HW = 1024 * 1024;                 // reference image plane size
    const int B  = in_sizes[0] / (3 * HW);      // batch count (8)

    const int threads = 256;
    const unsigned bx = (unsigned)((HW / 4 + threads - 1) / threads);
    dim3 grid(bx, (unsigned)B, 1);

    const size_t packed_bytes = (size_t)NLAT * sizeof(v4f); // 574,992 B

    if (ws_size >= packed_bytes) {
        // Repack LUT (tiny: ~1 MB of traffic), then apply.
        lut_repack_kernel<<<(NLAT + 255) / 256, 256, 0, stream>>>(
            lut, (v4f*)d_ws);
        lut_apply_packed_kernel<<<grid, threads, 0, stream>>>(
            x, (const v4f*)d_ws, out, HW);
    } else {
        lut_apply_direct_kernel<<<grid, threads, 0, stream>>>(
            x, lut, out, HW);
    }
}